// HANConv_64707977282160
// MI455X (gfx1250) — compile-verified
//
#include <hip/hip_runtime.h>

#define D_DIM 256

typedef float v2f __attribute__((ext_vector_type(2)));
typedef float v8f __attribute__((ext_vector_type(8)));

__device__ __forceinline__ v8f wmma_f32_4(v2f a, v2f b, v8f c) {
  // V_WMMA_F32_16X16X4_F32 : D = A(16x4,f32) * B(4x16,f32) + C(16x16,f32)
  return __builtin_amdgcn_wmma_f32_16x16x4_f32(false, a, false, b, (short)0, c,
                                               false, false);
}

// ---------------------------------------------------------------------------
// Repack W[256][256] row-major -> Wp[128][256] of float2 row-pairs:
//   Wp[p][c] = (W[2p][c], W[2p+1][c])
// so a B fragment (lane needs W[ka][col], W[ka+1][col]) is ONE b64 load and
// 16 consecutive lanes read 128B contiguous.
// ---------------------------------------------------------------------------
__global__ void han_pack_w(const float* __restrict__ W,
                           float2* __restrict__ Wp) {
  int idx = blockIdx.x * blockDim.x + threadIdx.x;  // 128*256 pairs
  if (idx >= 128 * 256) return;
  int p = idx >> 8;
  int c = idx & 255;
  Wp[idx] = make_float2(W[(size_t)(2 * p) * D_DIM + c],
                        W[(size_t)(2 * p + 1) * D_DIM + c]);
}

// ---------------------------------------------------------------------------
// C[16 x 256] strip = X[16 x 256] @ W[256 x 256] (+ bias), W in packed form.
// 128 threads = 4 waves; wave w owns output columns [64w, 64w+64) as 4 WMMA
// C tiles. A strip staged in LDS.
// ---------------------------------------------------------------------------
__global__ __launch_bounds__(128) void han_gemm_wmma(
    const float* __restrict__ X, const float2* __restrict__ Wp,
    const float* __restrict__ bias, float* __restrict__ out, int M) {
  __shared__ float As[16 * D_DIM];

  const int tid = threadIdx.x;
  const int lane = tid & 31;
  const int halfHi = lane >> 4;  // 0: lanes 0-15, 1: lanes 16-31
  const int l16 = lane & 15;
  const int wave = tid >> 5;
  const int row0 = blockIdx.x * 16;

  // Cooperative, coalesced load of the 16x256 A strip (zero-pad past M).
  const float4* Xv = (const float4*)(X + (size_t)row0 * D_DIM);
  float4* Asv = (float4*)As;
#pragma unroll
  for (int i = 0; i < 8; ++i) {
    int idx = tid + 128 * i;
    int r = idx >> 6;  // (idx*4)/256
    float4 v = make_float4(0.f, 0.f, 0.f, 0.f);
    if (row0 + r < M) v = Xv[idx];
    Asv[idx] = v;
  }
  __syncthreads();

  const int nbase = wave * 64;
  v8f acc[4] = {};

  for (int k = 0; k < D_DIM; k += 4) {
    const int ka = k + 2 * halfHi;  // ISA layout: lanes 16-31 hold K+2,K+3
    v2f a;
    a.x = As[l16 * D_DIM + ka];
    a.y = As[l16 * D_DIM + ka + 1];

    // packed pair row: lanes lo need pair k/2, lanes hi pair k/2+1
    const float2* Wrow =
        Wp + ((size_t)((k >> 1) + halfHi) << 8) + nbase + l16;
    if (k + 8 < D_DIM) __builtin_prefetch(Wrow + 4 * D_DIM, 0, 3);

    v2f b[4];
#pragma unroll
    for (int t = 0; t < 4; ++t) {
      float2 bb = Wrow[16 * t];  // global_load_b64, fully coalesced
      b[t].x = bb.x;
      b[t].y = bb.y;
    }
#pragma unroll
    for (int t = 0; t < 4; ++t) acc[t] = wmma_f32_4(a, b[t], acc[t]);
  }

  // C layout: VGPR r, lanes 0-15 -> (M=r, N=l16); lanes 16-31 -> (M=8+r, N=l16)
  const int rbase = halfHi * 8;
#pragma unroll
  for (int t = 0; t < 4; ++t) {
    const int col = nbase + 16 * t + l16;
    const float bv = bias ? bias[col] : 0.0f;
#pragma unroll
    for (int r = 0; r < 8; ++r) {
      int row = row0 + rbase + r;
      if (row < M) out[(size_t)row * D_DIM + col] = acc[t][r] + bv;
    }
  }
}

// ---------------------------------------------------------------------------
// score[n*2+off] = dot(tanh(H[n,:] @ Wsem + bsem), wsc)  (Wsem packed)
// Same WMMA body; epilogue fuses tanh + weighted column reduction so the
// [N,256] intermediate never hits memory.
// ---------------------------------------------------------------------------
__global__ __launch_bounds__(128) void han_score_wmma(
    const float* __restrict__ H, const float2* __restrict__ Wp,
    const float* __restrict__ bsem, const float* __restrict__ wsc,
    float* __restrict__ score, int off, int M) {
  __shared__ float As[16 * D_DIM];
  __shared__ float rowsum[16];

  const int tid = threadIdx.x;
  const int lane = tid & 31;
  const int halfHi = lane >> 4;
  const int l16 = lane & 15;
  const int wave = tid >> 5;
  const int row0 = blockIdx.x * 16;

  if (tid < 16) rowsum[tid] = 0.0f;

  const float4* Hv = (const float4*)(H + (size_t)row0 * D_DIM);
  float4* Asv = (float4*)As;
#pragma unroll
  for (int i = 0; i < 8; ++i) {
    int idx = tid + 128 * i;
    int r = idx >> 6;
    float4 v = make_float4(0.f, 0.f, 0.f, 0.f);
    if (row0 + r < M) v = Hv[idx];
    Asv[idx] = v;
  }
  __syncthreads();

  const int nbase = wave * 64;
  v8f acc[4] = {};

  for (int k = 0; k < D_DIM; k += 4) {
    const int ka = k + 2 * halfHi;
    v2f a;
    a.x = As[l16 * D_DIM + ka];
    a.y = As[l16 * D_DIM + ka + 1];
    const float2* Wrow =
        Wp + ((size_t)((k >> 1) + halfHi) << 8) + nbase + l16;
    v2f b[4];
#pragma unroll
    for (int t = 0; t < 4; ++t) {
      float2 bb = Wrow[16 * t];
      b[t].x = bb.x;
      b[t].y = bb.y;
    }
#pragma unroll
    for (int t = 0; t < 4; ++t) acc[t] = wmma_f32_4(a, b[t], acc[t]);
  }

  // per-lane partial row dot-products over this wave's 4 columns
  float pacc[8] = {0.f, 0.f, 0.f, 0.f, 0.f, 0.f, 0.f, 0.f};
#pragma unroll
  for (int t = 0; t < 4; ++t) {
    const int col = nbase + 16 * t + l16;
    const float bb = bsem[col];
    const float sw = wsc[col];
#pragma unroll
    for (int r = 0; r < 8; ++r) pacc[r] += tanhf(acc[t][r] + bb) * sw;
  }
  // reduce over the 16 lanes of each half-wave (rows 0-7 in lanes 0-15,
  // rows 8-15 in lanes 16-31); xor masks <= 8 stay within each half.
#pragma unroll
  for (int m = 1; m <= 8; m <<= 1) {
#pragma unroll
    for (int r = 0; r < 8; ++r) pacc[r] += __shfl_xor(pacc[r], m, 32);
  }
  if (l16 == 0) {
    const int rbase = halfHi * 8;
#pragma unroll
    for (int r = 0; r < 8; ++r)
      atomicAdd(&rowsum[rbase + r], pacc[r]);  // ds_add_f32
  }
  __syncthreads();
  if (tid < 16 && row0 + tid < M)
    score[(size_t)(row0 + tid) * 2 + off] = rowsum[tid];
}

// ---------------------------------------------------------------------------
// scatter-mean helpers
// ---------------------------------------------------------------------------
__global__ void han_deg(const int* __restrict__ col, float* __restrict__ deg,
                        int E) {
  int e = blockIdx.x * blockDim.x + threadIdx.x;
  if (e < E) atomicAdd(&deg[col[e]], 1.0f);
}

__global__ void han_scatter(const float* __restrict__ T,
                            const int* __restrict__ row,
                            const int* __restrict__ colv,
                            float* __restrict__ agg, int E) {
  int idx = blockIdx.x * blockDim.x + threadIdx.x;  // E * 64 float4-groups
  if (idx >= E * 64) return;
  int e = idx >> 6;
  int g = idx & 63;
  int r = row[e];
  int c = colv[e];
  const float4 v = *(const float4*)(T + (size_t)r * D_DIM + 4 * g);
  float* dst = agg + (size_t)c * D_DIM + 4 * g;
  atomicAdd(dst + 0, v.x);
  atomicAdd(dst + 1, v.y);
  atomicAdd(dst + 2, v.z);
  atomicAdd(dst + 3, v.w);
}

__global__ void han_div(float* __restrict__ agg, const float* __restrict__ deg,
                        int N) {
  int idx = blockIdx.x * blockDim.x + threadIdx.x;
  if (idx >= N * 64) return;
  int n = idx >> 6;
  int g = idx & 63;
  float inv = 1.0f / fmaxf(deg[n], 1.0f);
  float4* p = (float4*)(agg + (size_t)n * D_DIM + 4 * g);
  float4 v = *p;
  v.x *= inv; v.y *= inv; v.z *= inv; v.w *= inv;
  *p = v;
}

// 2-way softmax fuse; `out` may alias `h` (per-thread read-before-write).
__global__ void han_fuse(const float* h, const float* __restrict__ agg,
                         const float* __restrict__ score, float* out, int N) {
  int idx = blockIdx.x * blockDim.x + threadIdx.x;
  if (idx >= N * 64) return;
  int n = idx >> 6;
  int g = idx & 63;
  float s0 = score[2 * n];
  float s1 = score[2 * n + 1];
  float m = fmaxf(s0, s1);
  float e0 = __expf(s0 - m);
  float e1 = __expf(s1 - m);
  float inv = 1.0f / (e0 + e1);
  float w0 = e0 * inv, w1 = e1 * inv;
  const float4 a = *(const float4*)(h + (size_t)n * D_DIM + 4 * g);
  const float4 b = *(const float4*)(agg + (size_t)n * D_DIM + 4 * g);
  float4 o;
  o.x = w0 * a.x + w1 * b.x;
  o.y = w0 * a.y + w1 * b.y;
  o.z = w0 * a.z + w1 * b.z;
  o.w = w0 * a.w + w1 * b.w;
  *(float4*)(out + (size_t)n * D_DIM + 4 * g) = o;
}

extern "C" void kernel_launch(void* const* d_in, const int* in_sizes, int n_in,
                              void* d_out, int out_size, void* d_ws,
                              size_t ws_size, hipStream_t stream) {
  (void)n_in; (void)out_size; (void)ws_size;

  const float* x_author      = (const float*)d_in[0];
  const float* x_paper       = (const float*)d_in[1];
  const int*   row_writes    = (const int*)d_in[2];
  const int*   col_writes    = (const int*)d_in[3];
  const int*   row_written   = (const int*)d_in[4];
  const int*   col_written   = (const int*)d_in[5];
  const float* W_self_author = (const float*)d_in[6];
  const float* b_self_author = (const float*)d_in[7];
  const float* W_self_paper  = (const float*)d_in[8];
  const float* b_self_paper  = (const float*)d_in[9];
  const float* W_rel_writes  = (const float*)d_in[10];
  const float* W_rel_written = (const float*)d_in[11];
  const float* W_sem         = (const float*)d_in[12];
  const float* b_sem         = (const float*)d_in[13];
  const float* w_score       = (const float*)d_in[14];

  const int n_author = in_sizes[0] / D_DIM;
  const int n_paper  = in_sizes[1] / D_DIM;
  const int E        = in_sizes[2];

  const size_t NFa = (size_t)n_author * D_DIM;
  const size_t NFp = (size_t)n_paper * D_DIM;
  const size_t WSZ = (size_t)D_DIM * D_DIM;  // floats per weight matrix

  // d_out doubles as storage for the self-transform results (fused in place).
  float* h_author = (float*)d_out;
  float* h_paper  = (float*)d_out + NFa;

  float* ws           = (float*)d_ws;
  float* agg_paper    = ws;
  float* agg_author   = agg_paper + NFp;
  float* tmp          = agg_author + NFa;
  size_t tmpsz        = (NFa > NFp ? NFa : NFp);
  float* deg_paper    = tmp + tmpsz;
  float* deg_author   = deg_paper + n_paper;
  float* score_author = deg_author + n_author;
  float* score_paper  = score_author + 2 * (size_t)n_author;
  float* packbase     = score_paper + 2 * (size_t)n_paper;
  float2* wp_self_author = (float2*)(packbase + 0 * WSZ);
  float2* wp_self_paper  = (float2*)(packbase + 1 * WSZ);
  float2* wp_rel_writes  = (float2*)(packbase + 2 * WSZ);
  float2* wp_rel_written = (float2*)(packbase + 3 * WSZ);
  float2* wp_sem         = (float2*)(packbase + 4 * WSZ);

  // zero atomic accumulation targets (graph-capturable memset nodes)
  hipMemsetAsync(agg_paper, 0, (NFp + NFa) * sizeof(float), stream);
  hipMemsetAsync(deg_paper, 0, (size_t)(n_paper + n_author) * sizeof(float),
                 stream);

  // repack the five weight matrices into WMMA-fragment-native pair layout
  const unsigned gPack = (unsigned)((128 * 256 + 255) / 256);
  han_pack_w<<<gPack, 256, 0, stream>>>(W_self_author, wp_self_author);
  han_pack_w<<<gPack, 256, 0, stream>>>(W_self_paper, wp_self_paper);
  han_pack_w<<<gPack, 256, 0, stream>>>(W_rel_writes, wp_rel_writes);
  han_pack_w<<<gPack, 256, 0, stream>>>(W_rel_written, wp_rel_written);
  han_pack_w<<<gPack, 256, 0, stream>>>(W_sem, wp_sem);

  const dim3 blk(128);
  const unsigned strips_a = (unsigned)((n_author + 15) / 16);
  const unsigned strips_p = (unsigned)((n_paper + 15) / 16);
  const unsigned gE   = (unsigned)((E + 255) / 256);
  const unsigned gE64 = (unsigned)(((long long)E * 64 + 255) / 256);
  const unsigned gA64 = (unsigned)(((long long)n_author * 64 + 255) / 256);
  const unsigned gP64 = (unsigned)(((long long)n_paper * 64 + 255) / 256);

  // self transforms (directly into d_out)
  han_gemm_wmma<<<strips_a, blk, 0, stream>>>(x_author, wp_self_author,
                                              b_self_author, h_author, n_author);
  han_gemm_wmma<<<strips_p, blk, 0, stream>>>(x_paper, wp_self_paper,
                                              b_self_paper, h_paper, n_paper);

  // relation "writes": author -> paper
  han_gemm_wmma<<<strips_a, blk, 0, stream>>>(x_author, wp_rel_writes, nullptr,
                                              tmp, n_author);
  han_deg<<<gE, 256, 0, stream>>>(col_writes, deg_paper, E);
  han_scatter<<<gE64, 256, 0, stream>>>(tmp, row_writes, col_writes, agg_paper, E);
  han_div<<<gP64, 256, 0, stream>>>(agg_paper, deg_paper, n_paper);

  // relation "written_by": paper -> author
  han_gemm_wmma<<<strips_p, blk, 0, stream>>>(x_paper, wp_rel_written, nullptr,
                                              tmp, n_paper);
  han_deg<<<gE, 256, 0, stream>>>(col_written, deg_author, E);
  han_scatter<<<gE64, 256, 0, stream>>>(tmp, row_written, col_written,
                                        agg_author, E);
  han_div<<<gA64, 256, 0, stream>>>(agg_author, deg_author, n_author);

  // semantic attention scores (tanh GEMM fused with dot-reduction)
  han_score_wmma<<<strips_a, blk, 0, stream>>>(h_author, wp_sem, b_sem, w_score,
                                               score_author, 0, n_author);
  han_score_wmma<<<strips_a, blk, 0, stream>>>(agg_author, wp_sem, b_sem,
                                               w_score, score_author, 1,
                                               n_author);
  han_score_wmma<<<strips_p, blk, 0, stream>>>(h_paper, wp_sem, b_sem, w_score,
                                               score_paper, 0, n_paper);
  han_score_wmma<<<strips_p, blk, 0, stream>>>(agg_paper, wp_sem, b_sem,
                                               w_score, score_paper, 1,
                                               n_paper);

  // softmax fuse, in place over d_out halves
  han_fuse<<<gA64, 256, 0, stream>>>(h_author, agg_author, score_author,
                                     h_author, n_author);
  han_fuse<<<gP64, 256, 0, stream>>>(h_paper, agg_paper, score_paper, h_paper,
                                     n_paper);
}